// crossFrequencyAttention1_73005854098215
// MI455X (gfx1250) — compile-verified
//
#include <hip/hip_runtime.h>
#include <math.h>

// Problem constants (match reference)
#define B_DIM   16
#define H_DIM   4096
#define E_DIM   512
#define NM      24      // number of gathered frequency modes (index_q perm of 0..23)
#define NF      96      // QK feature columns: 24 complex Q (48 f32) + 24 complex K (48 f32)
#define KC      48      // K-feature columns (complex K modes flattened)
#define TWO_PI  6.28318530717958647692f

typedef float v2f __attribute__((ext_vector_type(2)));
typedef float v8f __attribute__((ext_vector_type(8)));
typedef int   v4i __attribute__((ext_vector_type(4)));

// CDNA5 async global->LDS copy (ASYNCcnt-tracked). Guarded: falls back to
// synchronous float4 staging if this toolchain lacks the builtins.
#if __has_builtin(__builtin_amdgcn_global_load_async_to_lds_b128)
#define HAVE_ASYNC_LDS 1
#else
#define HAVE_ASYNC_LDS 0
#endif

__device__ __forceinline__ void cp16_g2l(float* lds_dst, const float* gsrc) {
#if HAVE_ASYNC_LDS
  // prototype: (v4i addrspace(1)* gsrc, v4i addrspace(3)* ldsdst, imm offset, imm cpol)
  __builtin_amdgcn_global_load_async_to_lds_b128(
      (__attribute__((address_space(1))) v4i*)gsrc,
      (__attribute__((address_space(3))) v4i*)lds_dst, 0, 0);
#else
  *(float4*)lds_dst = *(const float4*)gsrc;
#endif
}

__device__ __forceinline__ void wait_async_lds() {
#if HAVE_ASYNC_LDS
#if __has_builtin(__builtin_amdgcn_s_wait_asynccnt)
  __builtin_amdgcn_s_wait_asynccnt(0);
#else
  asm volatile("s_wait_asynccnt 0x0" ::: "memory");
#endif
#endif
}

__device__ __forceinline__ v8f wmma_f32_16x16x4(v2f a, v2f b, v8f c) {
  // D = A(16x4 f32) * B(4x16 f32) + C(16x16 f32); wave32, VOP3P
  return __builtin_amdgcn_wmma_f32_16x16x4_f32(false, a, false, b, (short)0, c, false, false);
}

// ---------------------------------------------------------------------------
// P1: fold DFT-mode gather into the Q/K projection weights.
// Pair-interleaved output: Wcat2[(d/2)*96*2 + j*2 + (d&1)]  (d = K-dim row)
// so a WMMA B-fragment {k, k+1} is one aligned 8-byte LDS load.
// ---------------------------------------------------------------------------
__global__ void build_wcat(const float* __restrict__ Wq, const float* __restrict__ Wk,
                           const int* __restrict__ idxq, float* __restrict__ Wcat2) {
  const int d = blockIdx.x;          // 0..511 (K dimension of GEMM1)
  const int j = threadIdx.x;         // 0..95
  const float* W = (j < KC) ? Wq : Wk;
  const int jj = (j < KC) ? j : j - KC;
  const int m = jj >> 1, part = jj & 1;
  const int k = idxq[m];
  const float step = -TWO_PI * (float)k / (float)E_DIM;  // exp(-i theta)
  float acc = 0.0f;
  for (int e = 0; e < E_DIM; ++e) {
    float s, c;
    __sincosf(step * (float)e, &s, &c);
    const float w = W[(size_t)e * E_DIM + d];            // nn.Linear: x @ W.T
    acc += w * (part == 0 ? c : s);
  }
  Wcat2[(size_t)(d >> 1) * (NF * 2) + j * 2 + (d & 1)] = acc;
}

__global__ void build_bias(const float* __restrict__ bq, const float* __restrict__ bk,
                           const int* __restrict__ idxq, float* __restrict__ biascat) {
  const int j = threadIdx.x;         // 0..95
  const float* b = (j < KC) ? bq : bk;
  const int jj = (j < KC) ? j : j - KC;
  const int m = jj >> 1, part = jj & 1;
  const int k = idxq[m];
  const float step = -TWO_PI * (float)k / (float)E_DIM;
  float acc = 0.0f;
  for (int e = 0; e < E_DIM; ++e) {
    float s, c;
    __sincosf(step * (float)e, &s, &c);
    acc += b[e] * (part == 0 ? c : s);
  }
  biascat[j] = acc;
}

// ---------------------------------------------------------------------------
// P3: inverse rFFT basis restricted to the gathered modes.
// IF[2m][e] = s_k*cos(2pi k e/E)/E ; IF[2m+1][e] = -s_k*sin(.)/E ; s_k=(k==0?1:2)
// ---------------------------------------------------------------------------
__global__ void build_inv(const int* __restrict__ idxq, float* __restrict__ IF) {
  const int r = blockIdx.x;          // 0..47
  const int e = threadIdx.x;         // 0..511
  const int m = r >> 1, part = r & 1;
  const int k = idxq[m];
  const float scale = ((k == 0) ? 1.0f : 2.0f) / (float)E_DIM;
  float s, c;
  __sincosf(TWO_PI * (float)k * (float)e / (float)E_DIM, &s, &c);
  IF[(size_t)r * E_DIM + e] = (part == 0) ? scale * c : -scale * s;
}

__global__ void zero_buf(float* __restrict__ p, int n) {
  const int i = blockIdx.x * 256 + threadIdx.x;
  if (i < n) p[i] = 0.0f;
}

// ---------------------------------------------------------------------------
// GEMM1 (memory-bound, reads 128MB of x): QK[65536][96] = x @ Wcat + bias
// 256 thr = 8 waves, 128-row M tile, double-buffered async LDS pipeline.
// xs rows padded to 36 floats: keeps async B128 stores 16B-aligned AND makes
// the 16-lane A-frag reads conflict-free (36*r mod 64 distinct for r=0..15).
// ---------------------------------------------------------------------------
__global__ __launch_bounds__(256)
void gemm1_wmma(const float* __restrict__ x, const float* __restrict__ Wcat2,
                const float* __restrict__ biascat, float* __restrict__ QK) {
  __shared__ float xs[2][128][36];        // 2 x 18KB
  __shared__ float ws2[2][32 * NF];       // 2 x 12KB, pair-interleaved [16][96][2]
  const int t = threadIdx.x;
  const int wave = t >> 5, lane = t & 31;
  const int lh = lane >> 4, ll = lane & 15;
  const int row0 = blockIdx.x * 128;

  v8f acc[6];
  const v8f vzero = {0.f, 0.f, 0.f, 0.f, 0.f, 0.f, 0.f, 0.f};
#pragma unroll
  for (int n = 0; n < 6; ++n) acc[n] = vzero;

  // ---- stage tile k0 into buffer buf (async, 16B chunks) ----
  auto stage = [&](int buf, int k0) {
#pragma unroll
    for (int i = 0; i < 4; ++i) {          // x tile: 128x32 = 1024 float4
      const int idx = t + i * 256;
      const int r = idx >> 3, c4 = idx & 7;
      cp16_g2l(&xs[buf][r][c4 * 4],
               x + (size_t)(row0 + r) * E_DIM + k0 + c4 * 4);
    }
#pragma unroll
    for (int i = 0; i < 3; ++i) {          // W tile: 32*96 floats = 768 float4 (linear)
      const int idx = t + i * 256;
      cp16_g2l(&ws2[buf][idx * 4], Wcat2 + (size_t)k0 * NF + idx * 4);
    }
  };

  stage(0, 0);
  int buf = 0;
  for (int k0 = 0; k0 < E_DIM; k0 += 32) {
    wait_async_lds();
    __syncthreads();                       // tile `buf` resident, all waves past last compute
    if (k0 + 32 < E_DIM) stage(buf ^ 1, k0 + 32);
#pragma unroll
    for (int kk = 0; kk < 32; kk += 4) {
      const v2f a = *(const v2f*)&xs[buf][wave * 16 + ll][kk + 2 * lh];
      const int pr = (kk >> 1) + lh;       // K-pair row 0..15
#pragma unroll
      for (int n = 0; n < 6; ++n) {
        const v2f bf = *(const v2f*)&ws2[buf][pr * (NF * 2) + (n * 16 + ll) * 2];
        acc[n] = wmma_f32_16x16x4(a, bf, acc[n]);
      }
    }
    buf ^= 1;
  }
  // C layout: VGPR r, lanes0-15 -> M=r,N=lane ; lanes16-31 -> M=r+8,N=lane-16
#pragma unroll
  for (int n = 0; n < 6; ++n) {
    const int col = n * 16 + ll;
    const float bias = biascat[col];
#pragma unroll
    for (int r = 0; r < 8; ++r) {
      const int mrow = r + 8 * lh;
      QK[(size_t)(row0 + wave * 16 + mrow) * NF + col] = acc[n][r] + bias;
    }
  }
}

// ---------------------------------------------------------------------------
// Gram: S[b][x][y] (complex, no conj) = sum_h qf[b,h,x]*kf[b,h,y]; H split 8-way + f32 atomics
// ---------------------------------------------------------------------------
__global__ __launch_bounds__(576)
void gram_kernel(const float* __restrict__ QK, float* __restrict__ S) {
  __shared__ float tile[32][NF];
  const int b = blockIdx.x, chunk = blockIdx.y;
  const int t = threadIdx.x;                 // 0..575 == 24*24
  const int x = t / NM, y = t % NM;
  const int hbase = chunk * (H_DIM / 8);
  float accRe = 0.0f, accIm = 0.0f;
  for (int h0 = 0; h0 < H_DIM / 8; h0 += 32) {
    for (int idx = t; idx < 32 * NF; idx += 576) {
      const int r = idx / NF, c = idx % NF;
      tile[r][c] = QK[((size_t)b * H_DIM + hbase + h0 + r) * NF + c];
    }
    __syncthreads();
#pragma unroll 4
    for (int hh = 0; hh < 32; ++hh) {
      const float qr = tile[hh][2 * x],      qi = tile[hh][2 * x + 1];
      const float kr = tile[hh][KC + 2 * y], ki = tile[hh][KC + 2 * y + 1];
      accRe += qr * kr - qi * ki;
      accIm += qr * ki + qi * kr;
    }
    __syncthreads();
  }
  float* dst = S + (((size_t)b * NM + x) * NM + y) * 2;
  atomicAdd(dst, accRe);
  atomicAdd(dst + 1, accIm);
}

// ---------------------------------------------------------------------------
// softmax(|S|) over y, wave32 shuffles
// ---------------------------------------------------------------------------
__global__ void softmax_kernel(const float* __restrict__ S, float* __restrict__ att) {
  const int b = blockIdx.x / NM, x = blockIdx.x % NM;
  const int y = threadIdx.x;                 // 0..31
  const float* row = S + (((size_t)b * NM + x) * NM) * 2;
  float v = -1e30f;
  if (y < NM) {
    const float re = row[2 * y], im = row[2 * y + 1];
    v = sqrtf(re * re + im * im);
  }
  float m = v;
  for (int o = 16; o > 0; o >>= 1) m = fmaxf(m, __shfl_xor(m, o, 32));
  const float e = (y < NM) ? expf(v - m) : 0.0f;
  float s = e;
  for (int o = 16; o > 0; o >>= 1) s += __shfl_xor(s, o, 32);
  if (y < NM) att[((size_t)b * NM + x) * NM + y] = e / s;
}

// ---------------------------------------------------------------------------
// Fold att into inverse basis, pair-interleaved for GEMM2 B-fragments:
// Gp[((b*24 + r/2)*512 + e)*2 + (r&1)] = sum_x att[b,x,r/2] * IF[2x + (r&1)][e]
// ---------------------------------------------------------------------------
__global__ void gmat_kernel(const float* __restrict__ att, const float* __restrict__ IF,
                            float* __restrict__ Gp) {
  const int r = blockIdx.x;                  // 0..47
  const int b = blockIdx.y;
  const int e = threadIdx.x;                 // 0..511
  const int y = r >> 1, part = r & 1;
  float acc = 0.0f;
#pragma unroll
  for (int xm = 0; xm < NM; ++xm) {
    const float a = att[((size_t)b * NM + xm) * NM + y];
    acc += a * IF[(size_t)(2 * xm + part) * E_DIM + e];
  }
  Gp[(((size_t)b * NM + y) * E_DIM + e) * 2 + part] = acc;
}

// ---------------------------------------------------------------------------
// GEMM2 (writes 128MB): out[b,h][512] = Kft[b,h][48] @ G_b[48][512]
// 256 thr = 8 waves on a 32-row M tile; wave -> (m-half, 128-col N quarter).
// kf tile async-staged, rows padded to 52 floats (16B-aligned, conflict-free).
// G_b (96KB/batch, 1.5MB total) stays L2-hot; B-frag = one global b64 load.
// ---------------------------------------------------------------------------
__global__ __launch_bounds__(256)
void gemm2_wmma(const float* __restrict__ QK, const float* __restrict__ Gp,
                float* __restrict__ out) {
  __shared__ float kf[32][52];
  const int t = threadIdx.x;
  const int wave = t >> 5, lane = t & 31;
  const int lh = lane >> 4, ll = lane & 15;
  const int b = blockIdx.y;
  const int row0 = blockIdx.x * 32;
  const size_t bh0 = (size_t)b * H_DIM + row0;
  const float* __restrict__ Gpb = Gp + (size_t)b * NM * E_DIM * 2;

  // stage 32x48 K-feature tile (12 float4 per row, 384 chunks)
#pragma unroll
  for (int i = 0; i < 2; ++i) {
    const int idx = t + i * 256;
    if (idx < 384) {
      const int r = idx / 12, c4 = idx % 12;
      cp16_g2l(&kf[r][c4 * 4], QK + (bh0 + r) * NF + KC + c4 * 4);
    }
  }
  wait_async_lds();
  __syncthreads();

  const int mstrip = (wave & 1) * 16;
  const int ncol0 = (wave >> 1) * 128;
  v8f acc[8];
  const v8f vzero = {0.f, 0.f, 0.f, 0.f, 0.f, 0.f, 0.f, 0.f};
#pragma unroll
  for (int n = 0; n < 8; ++n) acc[n] = vzero;

#pragma unroll
  for (int kk = 0; kk < KC; kk += 4) {
    const v2f a = *(const v2f*)&kf[mstrip + ll][kk + 2 * lh];
    const int pr = (kk >> 1) + lh;           // K-pair row 0..23
#pragma unroll
    for (int n = 0; n < 8; ++n) {
      const int col = ncol0 + n * 16 + ll;
      const v2f bf = *(const v2f*)(Gpb + ((size_t)pr * E_DIM + col) * 2);
      acc[n] = wmma_f32_16x16x4(a, bf, acc[n]);
    }
  }
#pragma unroll
  for (int n = 0; n < 8; ++n) {
    const int col = ncol0 + n * 16 + ll;
#pragma unroll
    for (int r = 0; r < 8; ++r) {
      const int mrow = r + 8 * lh;
      out[(bh0 + mstrip + mrow) * E_DIM + col] = acc[n][r];
    }
  }
}

// ---------------------------------------------------------------------------
extern "C" void kernel_launch(void* const* d_in, const int* in_sizes, int n_in,
                              void* d_out, int out_size, void* d_ws, size_t ws_size,
                              hipStream_t stream) {
  (void)in_sizes; (void)n_in; (void)out_size; (void)ws_size;
  const float* x  = (const float*)d_in[0];
  const float* Wq = (const float*)d_in[1];
  const float* bq = (const float*)d_in[2];
  const float* Wk = (const float*)d_in[3];
  const float* bk = (const float*)d_in[4];
  // d_in[5]/d_in[6] = Wv/bv: dead in the reference (V unused) -> skipped entirely.
  const int* idxq = (const int*)d_in[7];

  // Workspace layout (floats); total ~27.1 MB
  float* ws      = (float*)d_ws;
  float* Wcat2   = ws;                                   // 512*96 (pair-interleaved)
  float* biascat = Wcat2 + (size_t)E_DIM * NF;           // 96
  float* IF      = biascat + NF;                         // 48*512
  float* QK      = IF + (size_t)KC * E_DIM;              // 65536*96
  float* S       = QK + (size_t)B_DIM * H_DIM * NF;      // 16*24*24*2
  float* att     = S + (size_t)B_DIM * NM * NM * 2;      // 16*24*24
  float* Gp      = att + (size_t)B_DIM * NM * NM;        // 16*48*512 (pair-interleaved)
  float* out     = (float*)d_out;

  build_wcat<<<E_DIM, NF, 0, stream>>>(Wq, Wk, idxq, Wcat2);
  build_bias<<<1, NF, 0, stream>>>(bq, bk, idxq, biascat);
  build_inv<<<KC, E_DIM, 0, stream>>>(idxq, IF);

  gemm1_wmma<<<(B_DIM * H_DIM) / 128, 256, 0, stream>>>(x, Wcat2, biascat, QK);

  const int s_elems = B_DIM * NM * NM * 2;
  zero_buf<<<(s_elems + 255) / 256, 256, 0, stream>>>(S, s_elems);
  gram_kernel<<<dim3(B_DIM, 8), 576, 0, stream>>>(QK, S);
  softmax_kernel<<<B_DIM * NM, 32, 0, stream>>>(S, att);
  gmat_kernel<<<dim3(KC, B_DIM), E_DIM, 0, stream>>>(att, IF, Gp);

  gemm2_wmma<<<dim3(H_DIM / 32, B_DIM), 256, 0, stream>>>(QK, Gp, out);
}